// Router_46119358824993
// MI455X (gfx1250) — compile-verified
//
#include <hip/hip_runtime.h>
#include <math.h>

#if __has_include(<hip/amd_detail/amd_gfx1250_TDM.h>)
#define ROUTER_TDM_6ARG 1
#endif

typedef __attribute__((ext_vector_type(2))) float        v2f;
typedef __attribute__((ext_vector_type(8))) float        v8f;
typedef __attribute__((ext_vector_type(4))) unsigned int v4u;
typedef __attribute__((ext_vector_type(4))) int          v4i;
typedef __attribute__((ext_vector_type(8))) int          v8i;

#define INPUT_DIM      2048
#define NUM_EXPERTS    64
#define TOKENS_TOTAL   16384
#define ROWS_PER_BLOCK 64            // 4 waves x 16-row WMMA tiles
#define WAVES          4
#define KCHUNK         64            // K staged per TDM tile
#define KSTEPS         (KCHUNK / 4)  // 16 WMMA K-steps per chunk
#define NTILES         (NUM_EXPERTS / 16)
#define NCHUNKS        (INPUT_DIM / KCHUNK)          // 32
#define XSTRIDE        68            // x LDS row: 64 data + 4 pad dwords (all 64 banks hit once)
#define WFRAG_FLOATS   (KSTEPS * NTILES * 32 * 2)    // 4096 floats = 16 KB per chunk

// top-2 insert with lax.top_k tie-break (equal value -> lower index wins)
#define TOP2_INSERT(vv, ii)                                                  \
    do {                                                                     \
        float _v = (vv); int _i = (ii);                                      \
        if (_v > v1 || (_v == v1 && _i < i1)) {                              \
            v2 = v1; i2 = i1; v1 = _v; i1 = _i;                              \
        } else if (_v > v2 || (_v == v2 && _i < i2)) {                       \
            v2 = _v; i2 = _i;                                                \
        }                                                                    \
    } while (0)

// ---- Tensor DMA descriptor helpers (CDNA5 ISA ch.8) ------------------------

__device__ __forceinline__ v8i tdm_group1(unsigned dim0, unsigned dim1,
                                          unsigned tile0, unsigned tile1,
                                          unsigned stride0, unsigned padEn,
                                          unsigned padIntervalEnc,
                                          unsigned padAmountEnc)
{
    v8i g1;
    // [15:0] wg_mask=0 | [17:16] data_size=2 (4B) | [20] pad_enable
    // [24:22] pad_interval | [31:25] pad_amount
    g1[0] = (int)((2u << 16) | (padEn << 20) | (padIntervalEnc << 22) | (padAmountEnc << 25));
    g1[1] = (int)((dim0 & 0xFFFFu) << 16);                              // tensor_dim0[15:0]
    g1[2] = (int)(((dim0 >> 16) & 0xFFFFu) | ((dim1 & 0xFFFFu) << 16)); // dim0 hi | dim1 lo
    g1[3] = (int)(((dim1 >> 16) & 0xFFFFu) | ((tile0 & 0xFFFFu) << 16));// dim1 hi | tile_dim0
    g1[4] = (int)(tile1 & 0xFFFFu);                                     // tile_dim1 | tile_dim2=0
    g1[5] = (int)stride0;                                               // tensor_dim0_stride lo
    g1[6] = 0;
    g1[7] = 0;
    return g1;
}

__device__ __forceinline__ void tdm_load_tile(unsigned lds_off, const void* gptr, v8i g1)
{
    unsigned long long ga = (unsigned long long)(size_t)gptr;
    v4u g0;
    g0.x = 1u;                                                   // count=1 (valid user D#)
    g0.y = lds_off;                                              // lds_addr (bytes)
    g0.z = (unsigned)ga;                                         // global_addr[31:0]
    g0.w = ((unsigned)(ga >> 32) & 0x01FFFFFFu) | (2u << 30);    // addr[56:32] | type=2
    v4i z4 = {0, 0, 0, 0};
#ifdef ROUTER_TDM_6ARG
    v8i z8 = {0, 0, 0, 0, 0, 0, 0, 0};
    __builtin_amdgcn_tensor_load_to_lds(g0, g1, z4, z4, z8, 0);
#else
    __builtin_amdgcn_tensor_load_to_lds(g0, g1, z4, z4, 0);
#endif
}

// ---- prep: pack W into WMMA-B fragment order (one-time, 512 KB) ------------
// wfrag[c][s][t][lane] = float2( W[kc+4s+2h][e], W[kc+4s+2h+1][e] ),
// h = lane>=16, e = (lane&15) + 16t  ->  compute loop does one ds_load_b64.
__global__ void pack_w_frags(const float* __restrict__ Wm, float* __restrict__ out)
{
    int id = blockIdx.x * blockDim.x + threadIdx.x;   // 0 .. 2048*64-1
    int k = id >> 6, e = id & 63;
    int c = k >> 6, kl = k & 63;
    int s = kl >> 2, kk = kl & 3;
    int h = kk >> 1, word = kk & 1;
    int t = e >> 4, p = (e & 15) + 16 * h;
    out[(size_t)c * WFRAG_FLOATS + (((s * NTILES + t) * 32 + p) << 1) + word] = Wm[id];
}

// ---------------------------------------------------------------------------

__launch_bounds__(128, 2)
__global__ void router_wmma_tdm(const float* __restrict__ x,
                                const float* __restrict__ wfrag,
                                const float* __restrict__ bvec,
                                int* __restrict__ idx_out,
                                float* __restrict__ gate_out)
{
    __shared__ float xLds[2][ROWS_PER_BLOCK * XSTRIDE];  // 2 x 17,408 B
    __shared__ float wLds[2][WFRAG_FLOATS];              // 2 x 16,384 B

    const int tid  = threadIdx.x;
    const int lane = tid & 31;
    const int wv   = tid >> 5;
    const int half = lane >> 4;          // 0: lanes 0-15 (K=0,1), 1: lanes 16-31 (K=2,3)
    const int col  = lane & 15;
    const int blockRow0 = blockIdx.x * ROWS_PER_BLOCK;
    const float* xBlk = x + (size_t)blockRow0 * INPUT_DIM;

    // x tile: 64 k (contig) x 64 rows; pad 64dw rows by 4dw -> LDS stride 68
    const v8i g1x = tdm_group1(INPUT_DIM, TOKENS_TOTAL, KCHUNK, ROWS_PER_BLOCK,
                               INPUT_DIM, 1, /*64dw*/5, /*4dw*/3);
    // W fragments: flat contiguous 4096-float block per chunk, no padding
    const v8i g1w = tdm_group1(WFRAG_FLOATS, NCHUNKS, WFRAG_FLOATS, 1,
                               WFRAG_FLOATS, 0, 0, 0);

    const unsigned xoff[2] = {(unsigned)(size_t)&xLds[0][0], (unsigned)(size_t)&xLds[1][0]};
    const unsigned woff[2] = {(unsigned)(size_t)&wLds[0][0], (unsigned)(size_t)&wLds[1][0]};

    v8f acc[NTILES];
#pragma unroll
    for (int t = 0; t < NTILES; ++t) acc[t] = (v8f){0, 0, 0, 0, 0, 0, 0, 0};

    if (wv == 0) {                       // prime the pipeline: chunk 0
        tdm_load_tile(xoff[0], xBlk, g1x);
        tdm_load_tile(woff[0], wfrag, g1w);
    }

    for (int c = 0; c < NCHUNKS; ++c) {
        if (wv == 0) {
            if (c + 1 < NCHUNKS) {       // issue chunk c+1 into the other buffer
                const int cn = c + 1;
                tdm_load_tile(xoff[cn & 1], xBlk + cn * KCHUNK, g1x);
                tdm_load_tile(woff[cn & 1], wfrag + (size_t)cn * WFRAG_FLOATS, g1w);
                __builtin_amdgcn_s_wait_tensorcnt(2);   // chunk c's 2 TDM ops done
            } else {
                __builtin_amdgcn_s_wait_tensorcnt(0);
            }
        }
        __syncthreads();                 // tile c visible to all waves

        const float*  xbuf  = xLds[c & 1];
        const float2* wbuf2 = (const float2*)wLds[c & 1];
        const int arow = (wv * 16 + col) * XSTRIDE + 2 * half;   // A: row-major + pad

#pragma unroll
        for (int s = 0; s < KSTEPS; ++s) {
            float2 a2 = *(const float2*)(xbuf + arow + 4 * s);   // ds_load_b64, bank-clean
            v2f av = {a2.x, a2.y};
#pragma unroll
            for (int t = 0; t < NTILES; ++t) {
                float2 b2 = wbuf2[(s * NTILES + t) * 32 + lane]; // ds_load_b64, pre-paired
                v2f bv = {b2.x, b2.y};
                acc[t] = __builtin_amdgcn_wmma_f32_16x16x4_f32(
                    false, av, false, bv, (short)0, acc[t], false, false);
            }
        }
        __syncthreads();                 // buffer c&1 free before chunk c+2 overwrites it
    }

    // ---- fused epilogue: bias + softmax + top-2, straight from accumulators ----
    float bias[NTILES];
#pragma unroll
    for (int t = 0; t < NTILES; ++t) bias[t] = bvec[col + 16 * t];

#pragma unroll
    for (int r = 0; r < 8; ++r) {
        // C VGPR r holds row r (lanes 0-15) and row r+8 (lanes 16-31); N = lane%16
        float v[NTILES];
        int   ei[NTILES];
#pragma unroll
        for (int t = 0; t < NTILES; ++t) {
            v[t]  = acc[t][r] + bias[t];
            ei[t] = col + 16 * t;
        }
        float v1 = v[0], v2 = -3.402823466e38f;
        int   i1 = ei[0], i2 = 0x7fffffff;
        TOP2_INSERT(v[1], ei[1]);
        TOP2_INSERT(v[2], ei[2]);
        TOP2_INSERT(v[3], ei[3]);
#pragma unroll
        for (int m = 1; m <= 8; m <<= 1) {   // 16-lane butterfly, stays in half-wave
            float w1 = __shfl_xor(v1, m, 32); int j1 = __shfl_xor(i1, m, 32);
            float w2 = __shfl_xor(v2, m, 32); int j2 = __shfl_xor(i2, m, 32);
            TOP2_INSERT(w1, j1);
            TOP2_INSERT(w2, j2);
        }
        float sl = 0.0f;                     // softmax denom; max == v1 after butterfly
#pragma unroll
        for (int t = 0; t < NTILES; ++t) sl += __expf(v[t] - v1);
#pragma unroll
        for (int m = 1; m <= 8; m <<= 1) sl += __shfl_xor(sl, m, 32);

        if (col == 0) {
            int token = blockRow0 + wv * 16 + r + 8 * half;
            float inv = 1.0f / sl;
            idx_out[2 * token]      = i1;
            idx_out[2 * token + 1]  = i2;
            gate_out[2 * token]     = inv;                      // exp(v1-v1)/S
            gate_out[2 * token + 1] = __expf(v2 - v1) * inv;
        }
    }
}

extern "C" void kernel_launch(void* const* d_in, const int* in_sizes, int n_in,
                              void* d_out, int out_size, void* d_ws, size_t ws_size,
                              hipStream_t stream) {
    const float* x  = (const float*)d_in[0];   // [4,4096,2048] f32
    const float* Wm = (const float*)d_in[1];   // [2048,64] f32
    const float* bv = (const float*)d_in[2];   // [64] f32

    const int tokens = in_sizes[0] / INPUT_DIM;          // 16384
    int*   idx_out  = (int*)d_out;                       // output 0: indices (int32)
    float* gate_out = (float*)((char*)d_out + (size_t)tokens * 2 * sizeof(int));
    float* wfrag    = (float*)d_ws;                      // 512 KB fragment-packed W

    // one-time (per launch) re-pack of W into WMMA-B fragment order
    pack_w_frags<<<(INPUT_DIM * NUM_EXPERTS) / 256, 256, 0, stream>>>(Wm, wfrag);

    const int blocks = tokens / ROWS_PER_BLOCK;          // 256
    router_wmma_tdm<<<blocks, 128, 0, stream>>>(x, wfrag, bv, idx_out, gate_out);
}